// Prgrph_ending_classifier_50835232916098
// MI455X (gfx1250) — compile-verified
//
#include <hip/hip_runtime.h>
#include <hip/hip_bf16.h>
#include <math.h>

typedef __attribute__((ext_vector_type(2))) float v2f;
typedef __attribute__((ext_vector_type(8))) float v8f;

#define S_DIM 50
#define E_DIM 256
#define N_ENT 20
#define LOG2_200 7.6438561897747395f
#define LDP 260   // padded LDS row stride: 260 % 64 == 4 -> conflict-free column gathers

// ---------------- kernel 0: positional embedding table (50x256) ----------------
__global__ void pos_kernel(float* __restrict__ pos) {
    int s = blockIdx.x;
    int e = threadIdx.x;
    // sin(s / 200^(e/E)) = sin(s * 2^(-(e/E)*log2(200)))
    float scale = exp2f(-((float)e / (float)E_DIM) * LOG2_200);
    pos[s * E_DIM + e] = __sinf((float)s * scale);
}

// ---------------- kernel 1: attention + entity sums (memory-bound pass) --------
__global__ void __launch_bounds__(256) stage1_kernel(
    const float* __restrict__ enc, const float* __restrict__ ent,
    const float* __restrict__ pos, float* __restrict__ ah, float* __restrict__ es)
{
    __shared__ float x[S_DIM * E_DIM];     // 51200 B
    __shared__ float logits[S_DIM];
    __shared__ float coef[S_DIM];

    const int b = blockIdx.x, tid = threadIdx.x;

    // x = enc[b] + pos   (coalesced float4 stream, read-once)
    const float4* enc4 = (const float4*)(enc + (size_t)b * S_DIM * E_DIM);
    const float4* pos4 = (const float4*)pos;
    float4* x4 = (float4*)x;
    for (int i = tid; i < S_DIM * E_DIM / 4; i += 256) {
        float4 a = enc4[i], p = pos4[i];
        x4[i] = make_float4(a.x + p.x, a.y + p.y, a.z + p.z, a.w + p.w);
    }
    __syncthreads();

    const int wave = tid >> 5, lane = tid & 31;

    // logits[s] = dot(x[49], x[s]) ; 8 waves split the 50 s-values
    float q[8];
    #pragma unroll
    for (int i = 0; i < 8; i++) q[i] = x[49 * E_DIM + lane + 32 * i];
    for (int s = wave; s < S_DIM; s += 8) {
        float acc = 0.f;
        #pragma unroll
        for (int i = 0; i < 8; i++) acc += q[i] * x[s * E_DIM + lane + 32 * i];
        #pragma unroll
        for (int off = 16; off > 0; off >>= 1) acc += __shfl_xor(acc, off, 32);
        if (lane == 0) logits[s] = acc;
    }
    __syncthreads();

    // softmax over 50 logits (wave 0 only)
    if (tid < 32) {
        float v0 = (lane < S_DIM)      ? logits[lane]      : -INFINITY;
        float v1 = (lane + 32 < S_DIM) ? logits[lane + 32] : -INFINITY;
        float m = fmaxf(v0, v1);
        #pragma unroll
        for (int off = 16; off > 0; off >>= 1) m = fmaxf(m, __shfl_xor(m, off, 32));
        float e0 = (lane < S_DIM)      ? __expf(v0 - m) : 0.f;
        float e1 = (lane + 32 < S_DIM) ? __expf(v1 - m) : 0.f;
        float sum = e0 + e1;
        #pragma unroll
        for (int off = 16; off > 0; off >>= 1) sum += __shfl_xor(sum, off, 32);
        float inv = 1.f / sum;
        if (lane < S_DIM)      coef[lane]      = e0 * inv;
        if (lane + 32 < S_DIM) coef[lane + 32] = e1 * inv;
    }
    __syncthreads();

    // attn_hidden[b, e] = sum_s coef[s] * x[s, e]   (LDS broadcast of coef)
    float acc = 0.f;
    for (int s = 0; s < S_DIM; s++) acc += coef[s] * x[s * E_DIM + tid];
    ah[(size_t)b * E_DIM + tid] = acc;

    // esum[b, e] = sum_n entities[b, n, e]   (coalesced read-once)
    float esum = 0.f;
    const float* eb = ent + (size_t)b * N_ENT * E_DIM;
    for (int n = 0; n < N_ENT; n++) esum += eb[n * E_DIM + tid];
    es[(size_t)b * E_DIM + tid] = esum;
}

// ---------------- kernel 2: WMMA GEMM (q_ent) + fused epilogue -----------------
// D = A(16xE) x attn(ExE); per block: 16 batch rows, 4 waves x 4 N-tiles of 16.
// V_WMMA_F32_16X16X4_F32 fragment layout (ISA 7.12.2):
//   A 16x4:  VGPR j, lane L<16: M=L, K=j ; lane L>=16: M=L-16, K=j+2
//   B 4x16:  VGPR j, lane L<16: K=j, N=L ; lane L>=16: K=j+2, N=L-16
//   D 16x16: VGPR v, lane L<16: M=v, N=L ; lane L>=16: M=v+8, N=L-16
__global__ void __launch_bounds__(128) stage2_kernel(
    const float* __restrict__ ah, const float* __restrict__ es,
    const float* __restrict__ attn, const float* __restrict__ w,
    const float* __restrict__ bias, float* __restrict__ out)
{
    __shared__ float ahl[16 * LDP];
    __shared__ float esl[16 * LDP];
    __shared__ float wl[E_DIM];
    __shared__ float rowacc[16];

    const int tid = threadIdx.x;
    const int m0 = blockIdx.x * 16;

    // stage A-panel and esum-panel with padded stride (bank-conflict-free gathers)
    const float4* ah4 = (const float4*)(ah + (size_t)m0 * E_DIM);
    const float4* es4 = (const float4*)(es + (size_t)m0 * E_DIM);
    for (int i = tid; i < 16 * E_DIM / 4; i += 128) {
        int row = i >> 6;           // 64 float4 per row
        int c4  = (i & 63) * 4;
        *(float4*)&ahl[row * LDP + c4] = ah4[i];
        *(float4*)&esl[row * LDP + c4] = es4[i];
    }
    wl[tid] = w[tid];
    wl[tid + 128] = w[tid + 128];
    if (tid < 16) rowacc[tid] = 0.f;
    __syncthreads();

    const int wave = tid >> 5, lane = tid & 31;
    const int r = lane & 15, hi = lane >> 4;

    v8f acc[4] = {};  // four 16x16 f32 accumulator tiles per wave
    for (int k0 = 0; k0 < E_DIM; k0 += 4) {
        v2f a;
        a.x = ahl[r * LDP + k0 + 2 * hi];       // A: M=r, K=k0+2*hi
        a.y = ahl[r * LDP + k0 + 2 * hi + 1];   //    M=r, K=k0+2*hi+1
        #pragma unroll
        for (int t = 0; t < 4; t++) {
            int n0 = (wave * 4 + t) * 16;
            v2f bm;
            bm.x = attn[(k0 + 2 * hi) * E_DIM + n0 + r];       // B: K, N=n0+r
            bm.y = attn[(k0 + 2 * hi + 1) * E_DIM + n0 + r];
            acc[t] = __builtin_amdgcn_wmma_f32_16x16x4_f32(
                false, a, false, bm, (short)0, acc[t], false, false);
        }
    }

    // epilogue: out_row[m] += q_ent[m, col] * esum[m, col] * w[col]
    #pragma unroll
    for (int v = 0; v < 8; v++) {
        int mrow = v + 8 * hi;
        float p = 0.f;
        #pragma unroll
        for (int t = 0; t < 4; t++) {
            int col = (wave * 4 + t) * 16 + r;
            p += acc[t][v] * esl[mrow * LDP + col] * wl[col];
        }
        // reduce over the 16 columns held by this half-wave (stay within hi group)
        #pragma unroll
        for (int off = 8; off > 0; off >>= 1) p += __shfl_xor(p, off, 32);
        if (r == 0) atomicAdd(&rowacc[mrow], p);   // ds_add_f32, 4 waves x 2 lanes
    }
    __syncthreads();

    if (tid < 16) {
        float z = rowacc[tid] + bias[0];
        out[m0 + tid] = 1.f / (1.f + __expf(-z));
    }
}

// ---------------- launch -------------------------------------------------------
extern "C" void kernel_launch(void* const* d_in, const int* in_sizes, int n_in,
                              void* d_out, int out_size, void* d_ws, size_t ws_size,
                              hipStream_t stream) {
    const float* enc  = (const float*)d_in[0];  // (B, 50, 256)
    const float* ent  = (const float*)d_in[1];  // (B, 20, 256)
    const float* attn = (const float*)d_in[2];  // (256, 256)
    const float* w    = (const float*)d_in[3];  // (256, 1)
    const float* bias = (const float*)d_in[4];  // (1,)
    float* out = (float*)d_out;                 // (B, 1) f32

    const int B = in_sizes[0] / (S_DIM * E_DIM);

    float* ws  = (float*)d_ws;
    float* pos = ws;                            // 50*256 floats
    float* ah  = ws + 16384;                    // B*256 floats (attn_hidden)
    float* es  = ah + (size_t)B * E_DIM;        // B*256 floats (entity sums)

    pos_kernel  <<<S_DIM, E_DIM, 0, stream>>>(pos);
    stage1_kernel<<<B,     256,  0, stream>>>(enc, ent, pos, ah, es);
    stage2_kernel<<<B / 16, 128, 0, stream>>>(ah, es, attn, w, bias, out);
}